// DegreeWiseLinear_13125420056867
// MI455X (gfx1250) — compile-verified
//
#include <hip/hip_runtime.h>

typedef float v2f __attribute__((ext_vector_type(2)));
typedef float v8f __attribute__((ext_vector_type(8)));

#define N_TOT   65536
#define ORD_IN  25
#define ORD_OUT 35
#define CH      64
#define N_EFF   24          // effective orders (input orders 1..24 -> output orders 0..23)
#define LDSTR   68          // 64 k-values + 4 pad: b64 stays 8B aligned, banks (4c+k) conflict-free

// One wave computes out[n0:n0+16, m, 0:64] = x[n0:n0+16, m+1, 0:64] @ W[deg(m)]
// using v_wmma_f32_16x16x4_f32 (exact fp32), K=64 in 16 chunks of 4.
__global__ __launch_bounds__(256)
void dwl_wmma_kernel(const float* __restrict__ x, const float* __restrict__ w,
                     float* __restrict__ out) {
    __shared__ float wT[CH * LDSTR];   // wT[c*LDSTR + k] = W[d][k][c] (transposed)

    const int m = blockIdx.x;                                   // 0..23
    const int d = (m < 3) ? 0 : (m < 8) ? 1 : (m < 15) ? 2 : 3; // degree index (l-1)

    // Cooperative transposed load of this degree's 64x64 weight into LDS.
    const float* wd = w + d * (CH * CH);
    for (int i = threadIdx.x; i < CH * CH; i += 256) {
        const int k = i >> 6, c = i & 63;
        wT[c * LDSTR + k] = wd[i];
    }
    __syncthreads();

    const int wave = threadIdx.x >> 5;
    const int lane = threadIdx.x & 31;
    const int n0   = (blockIdx.y * 8 + wave) * 16;

    const int half = lane >> 4;     // 0 -> K%4 in {0,1}; 1 -> K%4 in {2,3}
    const int l15  = lane & 15;     // A: row M;  B/D: column N (within 16-wide tile)
    const int koff = half << 1;

    // A fragment source: row (n0+l15), input order m+1, element (4*kc + koff .. +1)
    const float* aptr = x + (size_t)(n0 + l15) * (ORD_IN * CH) + (m + 1) * CH + koff;
    // B fragment source (LDS): column c = 16*ct + l15, k = 4*kc + koff .. +1
    const float* bptr = wT + l15 * LDSTR + koff;

    v8f acc0 = {}, acc1 = {}, acc2 = {}, acc3 = {};

    #pragma unroll
    for (int kc = 0; kc < 16; ++kc) {
        const v2f a  = *(const v2f*)(aptr + kc * 4);
        const v2f b0 = *(const v2f*)(bptr + kc * 4 + 0 * 16 * LDSTR);
        const v2f b1 = *(const v2f*)(bptr + kc * 4 + 1 * 16 * LDSTR);
        const v2f b2 = *(const v2f*)(bptr + kc * 4 + 2 * 16 * LDSTR);
        const v2f b3 = *(const v2f*)(bptr + kc * 4 + 3 * 16 * LDSTR);
        acc0 = __builtin_amdgcn_wmma_f32_16x16x4_f32(false, a, false, b0, (short)0, acc0, false, false);
        acc1 = __builtin_amdgcn_wmma_f32_16x16x4_f32(false, a, false, b1, (short)0, acc1, false, false);
        acc2 = __builtin_amdgcn_wmma_f32_16x16x4_f32(false, a, false, b2, (short)0, acc2, false, false);
        acc3 = __builtin_amdgcn_wmma_f32_16x16x4_f32(false, a, false, b3, (short)0, acc3, false, false);
    }

    // D layout: VGPR r -> M = r + 8*half, N = l15 (+16*ct)
    float* obase = out + (size_t)n0 * (ORD_OUT * CH) + m * CH;
    #pragma unroll
    for (int r = 0; r < 8; ++r) {
        float* orow = obase + (size_t)(half * 8 + r) * (ORD_OUT * CH) + l15;
        orow[0]  = acc0[r];
        orow[16] = acc1[r];
        orow[32] = acc2[r];
        orow[48] = acc3[r];
    }
}

// Zero the padded output tail: out[n, 24:35, 0:64] (contiguous 704 floats per n).
__global__ __launch_bounds__(256)
void dwl_zero_tail_kernel(float4* __restrict__ out4) {
    const int total = N_TOT * 176;                 // 704/4 float4 per row
    const int gid = blockIdx.x * 256 + threadIdx.x;
    if (gid < total) {
        const int n = gid / 176;
        const int j = gid - n * 176;
        // row stride = 35*64 floats = 560 float4; tail offset = 24*64 floats = 384 float4
        out4[(size_t)n * 560 + 384 + j] = make_float4(0.f, 0.f, 0.f, 0.f);
    }
}

extern "C" void kernel_launch(void* const* d_in, const int* in_sizes, int n_in,
                              void* d_out, int out_size, void* d_ws, size_t ws_size,
                              hipStream_t stream) {
    const float* x = (const float*)d_in[0];     // (65536, 25, 64) f32
    const float* w = (const float*)d_in[1];     // (4, 64, 64) f32
    float* out = (float*)d_out;                 // (65536, 35, 64) f32

    dim3 grid(N_EFF, N_TOT / (16 * 8));         // 24 x 512 blocks of 256 threads (8 waves)
    dwl_wmma_kernel<<<grid, 256, 0, stream>>>(x, w, out);

    const int ztotal = N_TOT * 176;
    dwl_zero_tail_kernel<<<(ztotal + 255) / 256, 256, 0, stream>>>((float4*)d_out);
}